// ChebNet_71940702208089
// MI455X (gfx1250) — compile-verified
//
#include <hip/hip_runtime.h>
#include <hip/hip_bf16.h>

typedef __attribute__((ext_vector_type(2))) float v2f;
typedef __attribute__((ext_vector_type(8))) float v8f;

#define FIN   50
#define FINP  52            // FIN padded to multiple of 4 (LDS stride, bank-conflict-free)
#define HID   32
#define HIDP  34            // HID padded for LDS bank spread
#define NCLS  10
#define NCLSP 16            // classes padded to one WMMA N-tile

// ---------------- utility kernels ----------------

__global__ void zero_f32(float* __restrict__ p, long long n) {
    long long i = (long long)blockIdx.x * blockDim.x + threadIdx.x;
    if (i < n) p[i] = 0.0f;
}

__global__ void deg_kernel(const int* __restrict__ src, float* __restrict__ deg, int E) {
    int e = blockIdx.x * blockDim.x + threadIdx.x;
    if (e < E) atomicAdd(&deg[src[e]], 1.0f);
}

__global__ void dinv_kernel(const float* __restrict__ deg, float* __restrict__ dinv, int N) {
    int i = blockIdx.x * blockDim.x + threadIdx.x;
    if (i < N) {
        float d = deg[i];
        dinv[i] = (d > 0.0f) ? rsqrtf(fmaxf(d, 1.0f)) : 0.0f;
    }
}

// pack each edge into one 16B record: {src, dst, bitcast(w), 0}
__global__ void pack_kernel(const int* __restrict__ src, const int* __restrict__ dst,
                            const float* __restrict__ dinv, int4* __restrict__ ep, int E) {
    int e = blockIdx.x * blockDim.x + threadIdx.x;
    if (e < E) {
        int s = src[e], d = dst[e];
        float w = -dinv[s] * dinv[d];
        ep[e] = make_int4(s, d, __float_as_int(w), 0);
    }
}

// gather feat[src]*w, scatter-add into out[dst]; one thread per (edge, feature);
// single b128 edge-record load (broadcast across the F threads sharing an edge)
template <int F>
__global__ void scatter_kernel(const float* __restrict__ feat, const int4* __restrict__ ep,
                               float* __restrict__ out, long long total) {
    long long tid = (long long)blockIdx.x * blockDim.x + threadIdx.x;
    if (tid >= total) return;
    int e = (int)(tid / F);
    int f = (int)(tid % F);
    int4 p = ep[e];
    float w = __int_as_float(p.z);
    atomicAdd(&out[(long long)p.y * F + f], w * feat[(long long)p.x * F + f]);
}

// ---------------- WMMA dense layers (fp32, 16x16x4) ----------------
// Weights staged in LDS TRANSPOSED (lW[col*stride + k]) so each B fragment
// (K-pair per lane) is one ds_load_b64; K zero-padded so the main loop has no guards.

__global__ __launch_bounds__(256)
void dense1_kernel(const float* __restrict__ x, const float* __restrict__ tx1,
                   const float* __restrict__ W1, const float* __restrict__ b1,
                   float* __restrict__ h, int N) {
    __shared__ float lW[2 * HID * FINP];            // 13312 B
    for (int i = threadIdx.x; i < 2 * HID * FINP; i += 256) {
        int m = i / (HID * FINP);
        int r = i % (HID * FINP);
        int c = r / FINP, k = r % FINP;
        lW[i] = (k < FIN) ? W1[m * FIN * HID + k * HID + c] : 0.0f;
    }
    __syncthreads();
    const float* lW0 = lW;
    const float* lW1 = lW + HID * FINP;

    const int lane  = threadIdx.x & 31;
    const int wave  = threadIdx.x >> 5;
    const int tile  = (blockIdx.x * 8 + wave) * 16;
    const int row   = tile + (lane & 15);
    const int rowC  = row < N ? row : N - 1;        // clamp: garbage rows never stored
    const int khalf = (lane >> 4) * 2;              // lanes 0-15 -> K{0,1}, 16-31 -> K{2,3}
    const int col   = lane & 15;

    const float* xr = x   + (long long)rowC * FIN;
    const float* tr = tx1 + (long long)rowC * FIN;

    v8f acc0 = {};  // cols 0..15
    v8f acc1 = {};  // cols 16..31

#pragma unroll
    for (int k = 0; k < 48; k += 4) {               // kk <= 47 < FIN: no guards
        int kk = k + khalf;
        v2f aX = *(const v2f*)(xr + kk);
        v2f aT = *(const v2f*)(tr + kk);
        v2f b0lo = *(const v2f*)(lW0 + col * FINP + kk);
        v2f b0hi = *(const v2f*)(lW0 + (col + 16) * FINP + kk);
        v2f b1lo = *(const v2f*)(lW1 + col * FINP + kk);
        v2f b1hi = *(const v2f*)(lW1 + (col + 16) * FINP + kk);
        acc0 = __builtin_amdgcn_wmma_f32_16x16x4_f32(false, aX, false, b0lo, (short)0, acc0, false, false);
        acc0 = __builtin_amdgcn_wmma_f32_16x16x4_f32(false, aT, false, b1lo, (short)0, acc0, false, false);
        acc1 = __builtin_amdgcn_wmma_f32_16x16x4_f32(false, aX, false, b0hi, (short)0, acc1, false, false);
        acc1 = __builtin_amdgcn_wmma_f32_16x16x4_f32(false, aT, false, b1hi, (short)0, acc1, false, false);
    }
    {   // tail K-step: kk in {48, 50}; LDS is zero-padded, A needs a guard
        int kk = 48 + khalf;
        v2f aX = {}, aT = {};
        if (kk < FIN) { aX = *(const v2f*)(xr + kk); aT = *(const v2f*)(tr + kk); }
        v2f b0lo = *(const v2f*)(lW0 + col * FINP + kk);
        v2f b0hi = *(const v2f*)(lW0 + (col + 16) * FINP + kk);
        v2f b1lo = *(const v2f*)(lW1 + col * FINP + kk);
        v2f b1hi = *(const v2f*)(lW1 + (col + 16) * FINP + kk);
        acc0 = __builtin_amdgcn_wmma_f32_16x16x4_f32(false, aX, false, b0lo, (short)0, acc0, false, false);
        acc0 = __builtin_amdgcn_wmma_f32_16x16x4_f32(false, aT, false, b1lo, (short)0, acc0, false, false);
        acc1 = __builtin_amdgcn_wmma_f32_16x16x4_f32(false, aX, false, b0hi, (short)0, acc1, false, false);
        acc1 = __builtin_amdgcn_wmma_f32_16x16x4_f32(false, aT, false, b1hi, (short)0, acc1, false, false);
    }

    // D layout: VGPR j -> M = tile + j + 8*(lane>>4), N = lane&15
    const int rbase = tile + ((lane >> 4) << 3);
    const float bias0 = b1[col], bias1 = b1[col + 16];
#pragma unroll
    for (int j = 0; j < 8; ++j) {
        int r = rbase + j;
        if (r < N) {
            float v0 = acc0[j] + bias0;
            float v1 = acc1[j] + bias1;
            h[(long long)r * HID + col]      = v0 > 0.0f ? v0 : 0.0f;
            h[(long long)r * HID + col + 16] = v1 > 0.0f ? v1 : 0.0f;
        }
    }
}

__global__ __launch_bounds__(256)
void dense2_kernel(const float* __restrict__ h, const float* __restrict__ tx2,
                   const float* __restrict__ W2, const float* __restrict__ b2,
                   float* __restrict__ logits, int N) {
    __shared__ float lW[2 * NCLSP * HIDP];          // 4352 B, cols 10..15 zero
    for (int i = threadIdx.x; i < 2 * NCLSP * HIDP; i += 256) {
        int m = i / (NCLSP * HIDP);
        int r = i % (NCLSP * HIDP);
        int c = r / HIDP, k = r % HIDP;
        lW[i] = (c < NCLS && k < HID) ? W2[m * HID * NCLS + k * NCLS + c] : 0.0f;
    }
    __syncthreads();
    const float* lW0 = lW;
    const float* lW1 = lW + NCLSP * HIDP;

    const int lane  = threadIdx.x & 31;
    const int wave  = threadIdx.x >> 5;
    const int tile  = (blockIdx.x * 8 + wave) * 16;
    const int row   = tile + (lane & 15);
    const int rowC  = row < N ? row : N - 1;
    const int khalf = (lane >> 4) * 2;
    const int col   = lane & 15;

    const float* hr = h   + (long long)rowC * HID;
    const float* tr = tx2 + (long long)rowC * HID;

    v8f acc = {};
#pragma unroll
    for (int k = 0; k < HID; k += 4) {              // K = 32 exact, no tail
        int kk = k + khalf;
        v2f aH = *(const v2f*)(hr + kk);
        v2f aT = *(const v2f*)(tr + kk);
        v2f b0 = *(const v2f*)(lW0 + col * HIDP + kk);
        v2f b1 = *(const v2f*)(lW1 + col * HIDP + kk);
        acc = __builtin_amdgcn_wmma_f32_16x16x4_f32(false, aH, false, b0, (short)0, acc, false, false);
        acc = __builtin_amdgcn_wmma_f32_16x16x4_f32(false, aT, false, b1, (short)0, acc, false, false);
    }

    const int rbase = tile + ((lane >> 4) << 3);
    const bool colOk = col < NCLS;
    const float bias = colOk ? b2[col] : 0.0f;
#pragma unroll
    for (int j = 0; j < 8; ++j) {
        int r = rbase + j;
        if (r < N && colOk) logits[(long long)r * NCLS + col] = acc[j] + bias;
    }
}

// ---------------- log-softmax over 10 classes ----------------
__global__ void logsoftmax_kernel(const float* __restrict__ logits, float* __restrict__ out, int N) {
    int i = blockIdx.x * blockDim.x + threadIdx.x;
    if (i >= N) return;
    float v[NCLS];
    float m = -INFINITY;
#pragma unroll
    for (int c = 0; c < NCLS; ++c) { v[c] = logits[(long long)i * NCLS + c]; m = fmaxf(m, v[c]); }
    float s = 0.0f;
#pragma unroll
    for (int c = 0; c < NCLS; ++c) s += __expf(v[c] - m);
    float lse = m + __logf(s);
#pragma unroll
    for (int c = 0; c < NCLS; ++c) out[(long long)i * NCLS + c] = v[c] - lse;
}

// ---------------- launch ----------------
extern "C" void kernel_launch(void* const* d_in, const int* in_sizes, int n_in,
                              void* d_out, int out_size, void* d_ws, size_t ws_size,
                              hipStream_t stream) {
    const float* x  = (const float*)d_in[0];
    const int*   ei = (const int*)d_in[1];     // edge_index (per harness integer mapping)
    const float* W1 = (const float*)d_in[2];   // (2, 50, 32)
    const float* b1 = (const float*)d_in[3];   // (32)
    const float* W2 = (const float*)d_in[4];   // (2, 32, 10)
    const float* b2 = (const float*)d_in[5];   // (10)

    const int N = in_sizes[0] / FIN;
    const int E = in_sizes[1] / 2;
    const int* srcI = ei;
    const int* dstI = ei + E;

    // workspace layout (floats); atomic targets first so one zero pass covers them.
    // N*(1+50+32) and +N are multiples of 4 -> epack is 16B-aligned off the base.
    float* ws   = (float*)d_ws;
    float* deg  = ws;                                  // N
    float* tx1  = deg + N;                             // N*FIN
    float* tx2  = tx1 + (size_t)N * FIN;               // N*HID
    float* dinv = tx2 + (size_t)N * HID;               // N
    int4*  ep   = (int4*)(dinv + N);                   // E records (16B each)
    float* h    = (float*)(ep + E);                    // N*HID
    float* lg   = h + (size_t)N * HID;                 // N*NCLS

    const int B = 256;
    long long zn = (long long)N * (1 + FIN + HID);
    zero_f32<<<(unsigned)((zn + B - 1) / B), B, 0, stream>>>(ws, zn);

    deg_kernel<<<(E + B - 1) / B, B, 0, stream>>>(srcI, deg, E);
    dinv_kernel<<<(N + B - 1) / B, B, 0, stream>>>(deg, dinv, N);
    pack_kernel<<<(E + B - 1) / B, B, 0, stream>>>(srcI, dstI, dinv, ep, E);

    long long t1 = (long long)E * FIN;
    scatter_kernel<FIN><<<(unsigned)((t1 + B - 1) / B), B, 0, stream>>>(x, ep, tx1, t1);

    int gDense = (N + 16 * 8 - 1) / (16 * 8);          // 8 waves/block, 16 rows/wave
    dense1_kernel<<<gDense, B, 0, stream>>>(x, tx1, W1, b1, h, N);

    long long t2 = (long long)E * HID;
    scatter_kernel<HID><<<(unsigned)((t2 + B - 1) / B), B, 0, stream>>>(h, ep, tx2, t2);

    dense2_kernel<<<gDense, B, 0, stream>>>(h, tx2, W2, b2, lg, N);

    logsoftmax_kernel<<<(N + B - 1) / B, B, 0, stream>>>(lg, (float*)d_out, N);
}